// ProposalTargetLayer_40467181863112
// MI455X (gfx1250) — compile-verified
//
#include <hip/hip_runtime.h>
#include <stdint.h>

#define TPB 256
#define MAX_G 256
#define NUM_ROIS_K 128
#define NUM_FG_MAX_K 32
#define BG_PID_K 5532

typedef unsigned int v4u __attribute__((ext_vector_type(4)));
typedef int v8i __attribute__((ext_vector_type(8)));
typedef int v4i __attribute__((ext_vector_type(4)));

#if __has_builtin(__builtin_amdgcn_tensor_load_to_lds) && __has_builtin(__builtin_amdgcn_s_wait_tensorcnt)
#define USE_TDM 1
#endif

__device__ __forceinline__ unsigned lds_byte_offset(const void* p) {
  // Shared aperture: low 32 bits of the flat address are the LDS byte offset.
  return (unsigned)(unsigned long long)(uintptr_t)p;
}

// ---------------------------------------------------------------------------
// Kernel 1: stage GT boxes into LDS via the Tensor Data Mover, compute per-ROI
// max/argmax IoU (division-free via cross-multiplied comparison), ballot-pack
// fg/bg masks (wave32), store argmax.
// ---------------------------------------------------------------------------
__global__ __launch_bounds__(TPB) void ptl_iou_kernel(
    const float* __restrict__ allRois, const float* __restrict__ gt,
    int R, int G,
    unsigned* __restrict__ fgMask, unsigned* __restrict__ bgMask,
    int* __restrict__ argmaxOut)
{
  __shared__ __align__(16) float sRaw[MAX_G * 6];   // raw gt_boxes [G,6]
  __shared__ __align__(16) float4 sBox[MAX_G];      // x1, y1, x2+1, y2+1
  __shared__ float sArea[MAX_G];                    // (w+1)*(h+1)

  const int tid = threadIdx.x;
  const int N = R + G;
  const int elems = G * 6;

#ifdef USE_TDM
  // One wave issues a 1-D TDM descriptor: 6144B gt table -> LDS.
  if ((tid >> 5) == 0) {
    unsigned long long ga = (unsigned long long)(uintptr_t)gt;
    unsigned ldsOff = lds_byte_offset(&sRaw[0]);
    // D# group0: count=1 | lds_addr | global_addr[56:0] | type=2
    v4u g0 = { 1u,
               ldsOff,
               (unsigned)(ga & 0xFFFFFFFFull),
               (unsigned)((ga >> 32) & 0x1FFFFFFull) | (2u << 30) };
    // D# group1: wg_mask=0, data_size=4B(enc 2), tensor_dim0=elems,
    // tensor_dim1=1, tile_dim0=elems, tile_dim1=1, stride0=stride1=elems
    v8i g1 = { (int)0x00020000u,
               (int)(((unsigned)elems & 0xFFFFu) << 16),
               (int)((((unsigned)elems >> 16) & 0xFFFFu) | (1u << 16)),
               (int)(((unsigned)elems & 0xFFFFu) << 16),
               1,
               elems,
               (int)(((unsigned)elems & 0xFFFFu) << 16),
               0 };
    // D# group2/3: higher dims = 1 / unused
    v4i g2 = { 1, 1, 0, 0 };
    v4i g3 = { 0, 0x00010000, 0, 0 };
    v8i g4 = { 0, 0, 0, 0, 0, 0, 0, 0 };   // extra group (6-arg toolchain form)
    __builtin_amdgcn_tensor_load_to_lds(g0, g1, g2, g3, g4, 0);
    __builtin_amdgcn_s_wait_tensorcnt(0);
  }
  __syncthreads();
#else
  for (int e = tid; e < elems; e += TPB) sRaw[e] = gt[e];
  __syncthreads();
#endif

  // Precompute SoA with the +1 convention folded in.
  for (int g = tid; g < G; g += TPB) {
    float x1 = sRaw[g * 6 + 0], y1 = sRaw[g * 6 + 1];
    float x2 = sRaw[g * 6 + 2], y2 = sRaw[g * 6 + 3];
    sBox[g] = make_float4(x1, y1, x2 + 1.0f, y2 + 1.0f);
    sArea[g] = (x2 - x1 + 1.0f) * (y2 - y1 + 1.0f);
  }
  __syncthreads();

  const int i = blockIdx.x * TPB + tid;
  const bool valid = i < N;

  float ax1 = 0.f, ay1 = 0.f, ax2p = 1.f, ay2p = 1.f, areaA = 1.f;
  if (valid) {
    if (i < R) {
      const float* p = allRois + (size_t)i * 5;
      ax1 = p[1]; ay1 = p[2];
      float ax2 = p[3], ay2 = p[4];
      ax2p = ax2 + 1.0f; ay2p = ay2 + 1.0f;
      areaA = (ax2 - ax1 + 1.0f) * (ay2 - ay1 + 1.0f);
    } else {
      int g = i - R;                 // appended gt_rois: box = gt box
      ax1 = sRaw[g * 6 + 0]; ay1 = sRaw[g * 6 + 1];
      float ax2 = sRaw[g * 6 + 2], ay2 = sRaw[g * 6 + 3];
      ax2p = ax2 + 1.0f; ay2p = ay2 + 1.0f;
      areaA = (ax2 - ax1 + 1.0f) * (ay2 - ay1 + 1.0f);
    }
  }

  // max/argmax of inter/den without per-pair division:
  // iou_g > iou_best  <=>  inter_g * den_best > inter_best * den_g  (den > 0)
  float bi = -1.0f, bd = 1.0f;
  int bestIdx = 0;
  #pragma unroll 4
  for (int g = 0; g < G; ++g) {
    float4 b = sBox[g];              // ds_load_b128 (broadcast)
    float iw = fminf(ax2p, b.z) - fmaxf(ax1, b.x);
    float ih = fminf(ay2p, b.w) - fmaxf(ay1, b.y);
    iw = fmaxf(iw, 0.0f);
    ih = fmaxf(ih, 0.0f);
    float inter = iw * ih;
    float den = areaA + sArea[g] - inter;
    bool better = inter * bd > bi * den;
    bi = better ? inter : bi;
    bd = better ? den : bd;
    bestIdx = better ? g : bestIdx;
  }

  bool fg = valid && (bi >= 0.5f * bd);                          // iou >= 0.5
  bool bg = valid && (bi < 0.5f * bd) && (bi >= 0.1f * bd);      // 0.1 <= iou < 0.5

#if __has_builtin(__builtin_amdgcn_ballot_w32)
  unsigned fm = __builtin_amdgcn_ballot_w32(fg);
  unsigned bm = __builtin_amdgcn_ballot_w32(bg);
#else
  unsigned fm = (unsigned)__ballot(fg);
  unsigned bm = (unsigned)__ballot(bg);
#endif
  if ((tid & 31) == 0 && i < N) {
    fgMask[i >> 5] = fm;
    bgMask[i >> 5] = bm;
  }
  if (valid) argmaxOut[i] = bestIdx;
}

// ---------------------------------------------------------------------------
// Kernel 2 (single block): two-level scan over bit-packed masks to realize the
// sequential "take first k" selection, then emit all 6 outputs for 128 slots.
// ---------------------------------------------------------------------------
__global__ __launch_bounds__(TPB) void ptl_select_kernel(
    const float* __restrict__ allRois, const float* __restrict__ gt,
    int R, int G,
    const unsigned* __restrict__ fgMask, const unsigned* __restrict__ bgMask,
    const int* __restrict__ argmaxArr,
    float* __restrict__ out)
{
  const int N = R + G;
  const int W = (N + 31) >> 5;

  __shared__ int cntF[TPB], cntB[TPB], offF[TPB], offB[TPB];
  __shared__ int keepLds[NUM_ROIS_K];
  __shared__ int sNumFg, sNumBg;

  const int t = threadIdx.x;
  const int perT = (W + TPB - 1) / TPB;
  const int w0 = t * perT;
  const int w1 = min(W, w0 + perT);

  int cf = 0, cb = 0;
  for (int w = w0; w < w1; ++w) {
    cf += __builtin_popcount(fgMask[w]);
    cb += __builtin_popcount(bgMask[w]);
  }
  cntF[t] = cf; cntB[t] = cb;
  if (t < NUM_ROIS_K) keepLds[t] = 0;   // keep defaults to index 0
  __syncthreads();

  if (t == 0) {
    int aF = 0, aB = 0;
    for (int j = 0; j < TPB; ++j) {
      offF[j] = aF; aF += cntF[j];
      offB[j] = aB; aB += cntB[j];
    }
    int numFg = min(NUM_FG_MAX_K, aF);
    int numBg = min(NUM_ROIS_K - numFg, aB);
    sNumFg = numFg; sNumBg = numBg;
  }
  __syncthreads();

  const int numFg = sNumFg, numBg = sNumBg;
  int rF = offF[t], rB = offB[t];
  for (int w = w0; w < w1; ++w) {
    unsigned fw = fgMask[w], bw = bgMask[w];
    while (fw) {
      int b = __builtin_ctz(fw); fw &= fw - 1;
      if (rF < numFg) keepLds[rF] = (w << 5) | b;
      ++rF;
    }
    while (bw) {
      int b = __builtin_ctz(bw); bw &= bw - 1;
      if (rB < numBg) keepLds[numFg + rB] = (w << 5) | b;
      ++rB;
    }
  }
  __syncthreads();

  if (t < NUM_ROIS_K) {
    const int k = keepLds[t];
    float batch, ex1, ey1, ex2, ey2;
    if (k < R) {
      const float* p = allRois + (size_t)k * 5;
      batch = p[0]; ex1 = p[1]; ey1 = p[2]; ex2 = p[3]; ey2 = p[4];
    } else {
      const float* p = gt + (size_t)(k - R) * 6;
      batch = 0.0f; ex1 = p[0]; ey1 = p[1]; ex2 = p[2]; ey2 = p[3];
    }

    // rois [128,5] at offset 0
    out[t * 5 + 0] = batch;
    out[t * 5 + 1] = ex1;
    out[t * 5 + 2] = ey1;
    out[t * 5 + 3] = ex2;
    out[t * 5 + 4] = ey2;

    const int a = argmaxArr[k];
    const float labF = gt[a * 6 + 4];
    const float pidF = gt[a * 6 + 5];
    const bool isFg = t < numFg;
    const float label = isFg ? labF : 0.0f;

    const int O_LAB = NUM_ROIS_K * 5;                 // 640
    const int O_PID = O_LAB + NUM_ROIS_K;             // 768
    const int O_BT  = O_PID + NUM_ROIS_K;             // 896
    const int O_BI  = O_BT + NUM_ROIS_K * 8;          // 1920
    const int O_BO  = O_BI + NUM_ROIS_K * 8;          // 2944

    ((int*)out)[O_LAB + t] = (int)label;                       // labels int32
    ((int*)out)[O_PID + t] = isFg ? (int)pidF : BG_PID_K;      // pids int32

    // bbox_transform vs matched gt, normalized by STDS (MEANS = 0)
    float gx1 = gt[a * 6 + 0], gy1 = gt[a * 6 + 1];
    float gx2 = gt[a * 6 + 2], gy2 = gt[a * 6 + 3];
    float ew = ex2 - ex1 + 1.0f, eh = ey2 - ey1 + 1.0f;
    float ecx = ex1 + 0.5f * ew, ecy = ey1 + 0.5f * eh;
    float gw = gx2 - gx1 + 1.0f, gh = gy2 - gy1 + 1.0f;
    float gcx = gx1 + 0.5f * gw, gcy = gy1 + 0.5f * gh;
    float tg0 = ((gcx - ecx) / ew) / 0.1f;
    float tg1 = ((gcy - ecy) / eh) / 0.1f;
    float tg2 = logf(gw / ew) / 0.2f;
    float tg3 = logf(gh / eh) / 0.2f;

    const bool gate = label > 0.0f;
    const int cls = (int)rintf(label);   // 0 or 1
    float bt[8], bi8[8], bo8[8];
    #pragma unroll
    for (int j = 0; j < 8; ++j) { bt[j] = 0.0f; bi8[j] = 0.0f; bo8[j] = 0.0f; }
    if (gate) {
      bt[cls * 4 + 0] = tg0; bt[cls * 4 + 1] = tg1;
      bt[cls * 4 + 2] = tg2; bt[cls * 4 + 3] = tg3;
      bi8[cls * 4 + 0] = 1.0f; bi8[cls * 4 + 1] = 1.0f;
      bi8[cls * 4 + 2] = 1.0f; bi8[cls * 4 + 3] = 1.0f;
      bo8[cls * 4 + 0] = 1.0f; bo8[cls * 4 + 1] = 1.0f;
      bo8[cls * 4 + 2] = 1.0f; bo8[cls * 4 + 3] = 1.0f;
    }
    #pragma unroll
    for (int j = 0; j < 8; ++j) {
      out[O_BT + t * 8 + j] = bt[j];
      out[O_BI + t * 8 + j] = bi8[j];
      out[O_BO + t * 8 + j] = bo8[j];
    }
  }
}

extern "C" void kernel_launch(void* const* d_in, const int* in_sizes, int n_in,
                              void* d_out, int out_size, void* d_ws, size_t ws_size,
                              hipStream_t stream) {
  (void)n_in; (void)out_size; (void)ws_size;
  const float* allRois = (const float*)d_in[0];
  const float* gt      = (const float*)d_in[1];
  const int R = in_sizes[0] / 5;   // 131072
  const int G = in_sizes[1] / 6;   // 256
  const int N = R + G;
  const int W = (N + 31) / 32;

  unsigned* fgMask = (unsigned*)d_ws;
  unsigned* bgMask = fgMask + W;
  int* argm = (int*)(bgMask + W);

  const int blocks = (N + TPB - 1) / TPB;
  ptl_iou_kernel<<<blocks, TPB, 0, stream>>>(allRois, gt, R, G, fgMask, bgMask, argm);
  ptl_select_kernel<<<1, TPB, 0, stream>>>(allRois, gt, R, G, fgMask, bgMask, argm,
                                           (float*)d_out);
}